// SAGEResBlock_51866025066812
// MI455X (gfx1250) — compile-verified
//
#include <hip/hip_runtime.h>
#include <hip/hip_bf16.h>

typedef __attribute__((ext_vector_type(2))) float v2f;
typedef __attribute__((ext_vector_type(8))) float v8f;

#define N_NODES 100000
#define N_EDGES 800000
#define D 96
#define BN_EPS 1e-5f
#define SLOPE 0.01f
#define LDSS 100  // padded LDS row stride (gcd(100,64-bank pattern) keeps 16 rows conflict-free, 8B aligned)

// ---------------------------------------------------------------------------
// Workspace layout (floats):
//   deg      [N_NODES]
//   agg_sum  [N_NODES*D]
//   agg_max  [N_NODES*D]   (uint-max trick; relu>=0 so bit order == float order, 0-init == empty)
//   hbuf     [N_NODES*D]   (hp, later reused for out2_raw)
//   stats    [4*D]         (sum1, sumsq1, sum2, sumsq2)
// ---------------------------------------------------------------------------

__global__ __launch_bounds__(256) void init_ws_kernel(float* __restrict__ ws, long long total) {
    long long i = (long long)blockIdx.x * blockDim.x + threadIdx.x;
    long long stride = (long long)gridDim.x * blockDim.x;
    for (; i < total; i += stride) ws[i] = 0.0f;
}

// hp = relu(x @ W_pool + b_pool)   -- one 16-row tile per block, 6 waves x one 16-col tile
__global__ __launch_bounds__(192) void hp_gemm_kernel(const float* __restrict__ x,
                                                      const float* __restrict__ Wp,
                                                      const float* __restrict__ bp,
                                                      float* __restrict__ hp) {
    __shared__ float xs[16 * LDSS];
    const int row0 = blockIdx.x * 16;
    const int tid = threadIdx.x;
    for (int i = tid; i < 16 * D; i += 192) {
        int r = i / D, c = i % D;
        xs[r * LDSS + c] = x[(long long)(row0 + r) * D + c];
    }
    __syncthreads();

    const int wave = tid / 32, lane = tid % 32;
    const int hi = lane >> 4;       // 0 for lanes 0-15, 1 for lanes 16-31
    const int nn = lane & 15;       // N within tile (B/C/D), also M for A
    const int n0 = wave * 16;
    v8f acc = {};
    #pragma unroll
    for (int k0 = 0; k0 < D; k0 += 4) {
        const int kb = k0 + 2 * hi;                 // A: K = 2*hi + vgpr ; B: same row pair
        v2f a, b;
        a.x = xs[nn * LDSS + kb];
        a.y = xs[nn * LDSS + kb + 1];
        b.x = Wp[kb * D + n0 + nn];
        b.y = Wp[(kb + 1) * D + n0 + nn];
        acc = __builtin_amdgcn_wmma_f32_16x16x4_f32(false, a, false, b, (short)0, acc,
                                                    false, false);
    }
    const float bias = bp[n0 + nn];
    #pragma unroll
    for (int r = 0; r < 8; ++r) {
        float v = acc[r] + bias;
        v = v > 0.0f ? v : 0.0f;                     // relu
        hp[(long long)(row0 + r + 8 * hi) * D + n0 + nn] = v;
    }
}

// One wave per edge; each lane handles 3 channels. atomic add for mean-agg, uint atomic max for pool-agg.
__global__ __launch_bounds__(256) void edge_scatter_kernel(const long long* __restrict__ ei,
                                                           const float* __restrict__ x,
                                                           const float* __restrict__ hp,
                                                           float* __restrict__ deg,
                                                           float* __restrict__ asum,
                                                           unsigned int* __restrict__ amax) {
    long long tid = (long long)blockIdx.x * 256 + threadIdx.x;
    int e = (int)(tid >> 5);
    int lane = (int)(tid & 31);
    if (e >= N_EDGES) return;
    const int src = (int)ei[e];
    const int dst = (int)ei[N_EDGES + e];
    if (lane == 0) atomicAdd(&deg[dst], 1.0f);
    const float* xs = x + (long long)src * D;
    const float* hs = hp + (long long)src * D;
    float* ad = asum + (long long)dst * D;
    unsigned int* am = amax + (long long)dst * D;
    #pragma unroll
    for (int j = 0; j < 3; ++j) {
        const int c = lane + 32 * j;
        atomicAdd(&ad[c], xs[c]);
        atomicMax(&am[c], __float_as_uint(hs[c]));   // hp >= 0 -> bit order == value order
    }
}

// out1 = x@Ws1 + (asum/max(deg,1))@Wn1 ;  out2 = x@Ws2 + amax@Wn2
__global__ __launch_bounds__(192) void main_gemm_kernel(const float* __restrict__ x,
                                                        const float* __restrict__ deg,
                                                        const float* __restrict__ asum,
                                                        const float* __restrict__ amax,
                                                        const float* __restrict__ Ws1,
                                                        const float* __restrict__ Wn1,
                                                        const float* __restrict__ Ws2,
                                                        const float* __restrict__ Wn2,
                                                        float* __restrict__ out1,
                                                        float* __restrict__ out2) {
    __shared__ float xs[16 * LDSS];
    __shared__ float hm[16 * LDSS];
    __shared__ float hx[16 * LDSS];
    const int row0 = blockIdx.x * 16;
    const int tid = threadIdx.x;
    for (int i = tid; i < 16 * D; i += 192) {
        int r = i / D, c = i % D;
        long long g = (long long)(row0 + r) * D + c;
        xs[r * LDSS + c] = x[g];
        hm[r * LDSS + c] = asum[g] / fmaxf(deg[row0 + r], 1.0f);
        hx[r * LDSS + c] = amax[g];                  // already 0 for isolated nodes
    }
    __syncthreads();

    const int wave = tid / 32, lane = tid % 32;
    const int hi = lane >> 4;
    const int nn = lane & 15;
    const int n0 = wave * 16;
    v8f acc1 = {}, acc2 = {};
    #pragma unroll
    for (int k0 = 0; k0 < D; k0 += 4) {
        const int kb = k0 + 2 * hi;
        v2f ax, ah, am2, b;
        ax.x = xs[nn * LDSS + kb];  ax.y = xs[nn * LDSS + kb + 1];
        ah.x = hm[nn * LDSS + kb];  ah.y = hm[nn * LDSS + kb + 1];
        am2.x = hx[nn * LDSS + kb]; am2.y = hx[nn * LDSS + kb + 1];

        b.x = Ws1[kb * D + n0 + nn]; b.y = Ws1[(kb + 1) * D + n0 + nn];
        acc1 = __builtin_amdgcn_wmma_f32_16x16x4_f32(false, ax, false, b, (short)0, acc1, false, false);
        b.x = Wn1[kb * D + n0 + nn]; b.y = Wn1[(kb + 1) * D + n0 + nn];
        acc1 = __builtin_amdgcn_wmma_f32_16x16x4_f32(false, ah, false, b, (short)0, acc1, false, false);
        b.x = Ws2[kb * D + n0 + nn]; b.y = Ws2[(kb + 1) * D + n0 + nn];
        acc2 = __builtin_amdgcn_wmma_f32_16x16x4_f32(false, ax, false, b, (short)0, acc2, false, false);
        b.x = Wn2[kb * D + n0 + nn]; b.y = Wn2[(kb + 1) * D + n0 + nn];
        acc2 = __builtin_amdgcn_wmma_f32_16x16x4_f32(false, am2, false, b, (short)0, acc2, false, false);
    }
    #pragma unroll
    for (int r = 0; r < 8; ++r) {
        long long g = (long long)(row0 + r + 8 * hi) * D + n0 + nn;
        out1[g] = acc1[r];
        out2[g] = acc2[r];
    }
}

// Per-channel sums / sums-of-squares for both BN branches. One channel per thread (coalesced rows).
__global__ __launch_bounds__(96) void bn_stats_kernel(const float* __restrict__ o1,
                                                      const float* __restrict__ o2,
                                                      float* __restrict__ stats) {
    const int c = threadIdx.x;
    float s1 = 0.f, q1 = 0.f, s2 = 0.f, q2 = 0.f;
    for (int row = blockIdx.x; row < N_NODES; row += gridDim.x) {
        const float a = o1[(long long)row * D + c];
        const float b = o2[(long long)row * D + c];
        s1 += a; q1 += a * a;
        s2 += b; q2 += b * b;
    }
    atomicAdd(&stats[c],           s1);
    atomicAdd(&stats[D + c],       q1);
    atomicAdd(&stats[2 * D + c],   s2);
    atomicAdd(&stats[3 * D + c],   q2);
}

// BN(out1) + BN(out2) -> leaky relu, in place on d_out.
__global__ __launch_bounds__(256) void finalize_kernel(float* __restrict__ out,
                                                       const float* __restrict__ o2,
                                                       const float* __restrict__ stats,
                                                       const float* __restrict__ g1,
                                                       const float* __restrict__ be1,
                                                       const float* __restrict__ g2,
                                                       const float* __restrict__ be2) {
    const long long total = (long long)N_NODES * D;
    const float invN = 1.0f / (float)N_NODES;
    long long i = (long long)blockIdx.x * 256 + threadIdx.x;
    const long long stride = (long long)gridDim.x * 256;
    for (; i < total; i += stride) {
        const int c = (int)(i % D);
        const float mu1 = stats[c] * invN;
        const float var1 = stats[D + c] * invN - mu1 * mu1;
        const float mu2 = stats[2 * D + c] * invN;
        const float var2 = stats[3 * D + c] * invN - mu2 * mu2;
        float v = g1[c] * (out[i] - mu1) * rsqrtf(var1 + BN_EPS) + be1[c]
                + g2[c] * (o2[i]  - mu2) * rsqrtf(var2 + BN_EPS) + be2[c];
        out[i] = v > 0.0f ? v : SLOPE * v;
    }
}

extern "C" void kernel_launch(void* const* d_in, const int* in_sizes, int n_in,
                              void* d_out, int out_size, void* d_ws, size_t ws_size,
                              hipStream_t stream) {
    const float*     x   = (const float*)d_in[0];
    const long long* ei  = (const long long*)d_in[1];   // int64 edge_index [2, E]
    const float*     Ws1 = (const float*)d_in[2];
    const float*     Wn1 = (const float*)d_in[3];
    const float*     Wp  = (const float*)d_in[4];
    const float*     bp  = (const float*)d_in[5];
    const float*     Ws2 = (const float*)d_in[6];
    const float*     Wn2 = (const float*)d_in[7];
    const float*     g1  = (const float*)d_in[8];
    const float*     be1 = (const float*)d_in[9];
    const float*     g2  = (const float*)d_in[10];
    const float*     be2 = (const float*)d_in[11];
    float* out = (float*)d_out;

    float* ws    = (float*)d_ws;
    float* deg   = ws;
    float* asum  = deg  + N_NODES;
    float* amax  = asum + (size_t)N_NODES * D;
    float* hbuf  = amax + (size_t)N_NODES * D;     // hp, then reused as out2_raw
    float* stats = hbuf + (size_t)N_NODES * D;
    const long long total_ws = (long long)N_NODES + 3LL * N_NODES * D + 4 * D;

    // 1. zero deg / agg_sum / agg_max(0-bits == empty) / stats (hbuf overwritten by GEMM)
    init_ws_kernel<<<2048, 256, 0, stream>>>(ws, total_ws);

    // 2. hp = relu(x @ W_pool + b_pool)
    hp_gemm_kernel<<<N_NODES / 16, 192, 0, stream>>>(x, Wp, bp, hbuf);

    // 3. edge scatter: deg count, sum-agg of x, max-agg of hp
    edge_scatter_kernel<<<(N_EDGES * 32) / 256, 256, 0, stream>>>(
        ei, x, hbuf, deg, asum, (unsigned int*)amax);

    // 4. fused 4-GEMM: out1_raw -> d_out, out2_raw -> hbuf (hp no longer needed)
    main_gemm_kernel<<<N_NODES / 16, 192, 0, stream>>>(
        x, deg, asum, amax, Ws1, Wn1, Ws2, Wn2, out, hbuf);

    // 5. BN batch statistics for both branches
    bn_stats_kernel<<<256, 96, 0, stream>>>(out, hbuf, stats);

    // 6. BN + residual add + leaky relu
    finalize_kernel<<<4096, 256, 0, stream>>>(out, hbuf, stats, g1, be1, g2, be2);
}